// ddpOptimizer_78417512891186
// MI455X (gfx1250) — compile-verified
//
#include <hip/hip_runtime.h>

#if defined(__AMDGCN__) && __has_builtin(__builtin_amdgcn_global_load_async_to_lds_b128) && \
    __has_builtin(__builtin_amdgcn_global_load_async_to_lds_b32) && \
    __has_builtin(__builtin_amdgcn_s_wait_asynccnt)
#define USE_ASYNC 1
#else
#define USE_ASYNC 0
#endif

namespace {
constexpr int B_ = 256, T_ = 128, NS = 12, NC = 4, MM = 6;
constexpr float MU = 0.1f;
constexpr float REG_FAC = 1.6f * 1.6f - 1.0f;   // REG_BASE**REG - 1
constexpr int NT = 128;                          // 4 waves cooperate on one batch

// ---- output offsets (floats), concatenated in reference return order ----
constexpr size_t O_KU  = 0;                                   // ku  (B,T,NC)
constexpr size_t O_KUM = O_KU  + (size_t)B_ * T_ * NC;        // Ku  (B,T,NC,NS)
constexpr size_t O_KS  = O_KUM + (size_t)B_ * T_ * NC * NS;   // ks  (B,T,M)
constexpr size_t O_KSM = O_KS  + (size_t)B_ * T_ * MM;        // Ks  (B,T,M,NS)
constexpr size_t O_DV  = O_KSM + (size_t)B_ * T_ * MM * NS;   // dV  (B,2)
constexpr size_t O_OPT = O_DV  + (size_t)B_ * 2;              // opterr scalar
constexpr size_t O_VX  = O_OPT + 1;                           // Vx  (B,NS)
constexpr size_t O_VXX = O_VX  + (size_t)B_ * NS;             // Vxx (B,NS,NS)

// ---- input-slice offsets relative to a staging buffer (floats) ----
constexpr int RFX = 0, RFU = 144, RCX = 192, RCU = 264, RQXX = 288, RQXU = 432,
  RQUU = 480, RQX = 496, RQU = 512, RC = 528, RS = 536, RFXX = 544,
  RFXU = 2272, RFUU = 2848, BUFSZ = 3040;       // double-buffered: [0,6080)

// ---- scratch LDS offsets (floats) ----
constexpr int LVXX = 6080, LWT = 6336, LZT = 6592, LVX = 6848, LQXV = 6864,
  LQUV = 6880, LRR = 6888, LCINV = 6896, LSC = 6904, LCUSC = 6912,
  LQXXA = 6928, LQXUA = 7072, LQUUR = 7120, LQUUS = 7136, LG = 7152,
  LQUU2 = 7168, LQU2 = 7184, LTQUX = 7192, LQXX2 = 7240, LQX2 = 7384,
  LKK = 7400, LKTQ = 7464, LQK = 7512, LVXN = 7656, LVN = 7672;
constexpr int LDSF = 7816;                      // 31.3 KB per block
} // namespace

typedef __attribute__((ext_vector_type(2))) float v2f;
typedef __attribute__((ext_vector_type(8))) float v8f;
typedef __attribute__((ext_vector_type(4))) int v4i;

// stage one (b,t) input slice into LDS: async DMA (ASYNCcnt) when available,
// else coalesced sync loads + ds stores.
__device__ inline void stage(float* dst, const float* src, int n, int tid) {
#if USE_ASYNC
  if ((n & 3) == 0) {
    const int n4 = n >> 2;
    for (int i = tid; i < n4; i += NT)
      __builtin_amdgcn_global_load_async_to_lds_b128(
          (__attribute__((address_space(1))) v4i*)(const_cast<float*>(src) + 4 * i),
          (__attribute__((address_space(3))) v4i*)(dst + 4 * i), 0, 0);
  } else {
    for (int i = tid; i < n; i += NT)
      __builtin_amdgcn_global_load_async_to_lds_b32(
          (__attribute__((address_space(1))) int*)(const_cast<float*>(src) + i),
          (__attribute__((address_space(3))) int*)(dst + i), 0, 0);
  }
#else
  if ((n & 3) == 0) {
    const float4* s4 = reinterpret_cast<const float4*>(src);
    float4* d4 = reinterpret_cast<float4*>(dst);
    const int n4 = n >> 2;
    for (int i = tid; i < n4; i += NT) d4[i] = s4[i];
  } else {
    for (int i = tid; i < n; i += NT) dst[i] = src[i];
  }
#endif
}

__global__ void ddp_init_kernel(float* out) {
  if (threadIdx.x == 0 && blockIdx.x == 0) out[O_OPT] = 0.0f;
}

__global__ __launch_bounds__(NT) void ddp_bwd_kernel(
    const float* __restrict__ g_qx,  const float* __restrict__ g_qu,
    const float* __restrict__ g_qxx, const float* __restrict__ g_qxu,
    const float* __restrict__ g_quu, const float* __restrict__ g_fx,
    const float* __restrict__ g_fu,  const float* __restrict__ g_fxx,
    const float* __restrict__ g_fxu, const float* __restrict__ g_fuu,
    const float* __restrict__ g_cx,  const float* __restrict__ g_cu,
    const float* __restrict__ g_c,   const float* __restrict__ g_s,
    const float* __restrict__ g_px,  const float* __restrict__ g_pxx,
    float* __restrict__ out) {
  __shared__ __align__(16) float L[LDSF];
  const int tid  = threadIdx.x;
  const int lane = tid & 31;
  const int wv   = tid >> 5;
  const int b    = blockIdx.x;

  // init carry: Vxx = pxx (16x16 zero-padded tile), Vx = px
  for (int i = tid; i < 256; i += NT) {
    const int r = i >> 4, n = i & 15;
    L[LVXX + i] = (r < NS && n < NS) ? g_pxx[(size_t)b * NS * NS + r * NS + n] : 0.0f;
  }
  for (int i = tid; i < 16; i += NT)
    L[LVX + i] = (i < NS) ? g_px[(size_t)b * NS + i] : 0.0f;

  // helper to stage all 14 arrays of one (b,t) slice into a buffer
  auto stage_step = [&](float* buf, size_t bt) {
    stage(buf + RQX,  g_qx  + bt * NS,           NS,           tid);
    stage(buf + RQU,  g_qu  + bt * NC,           NC,           tid);
    stage(buf + RQXX, g_qxx + bt * NS * NS,      NS * NS,      tid);
    stage(buf + RQXU, g_qxu + bt * NS * NC,      NS * NC,      tid);
    stage(buf + RQUU, g_quu + bt * NC * NC,      NC * NC,      tid);
    stage(buf + RFX,  g_fx  + bt * NS * NS,      NS * NS,      tid);
    stage(buf + RFU,  g_fu  + bt * NS * NC,      NS * NC,      tid);
    stage(buf + RFXX, g_fxx + bt * NS * NS * NS, NS * NS * NS, tid);
    stage(buf + RFXU, g_fxu + bt * NS * NS * NC, NS * NS * NC, tid);
    stage(buf + RFUU, g_fuu + bt * NS * NC * NC, NS * NC * NC, tid);
    stage(buf + RCX,  g_cx  + bt * MM * NS,      MM * NS,      tid);
    stage(buf + RCU,  g_cu  + bt * MM * NC,      MM * NC,      tid);
    stage(buf + RC,   g_c   + bt * MM,           MM,           tid);
    stage(buf + RS,   g_s   + bt * MM,           MM,           tid);
  };

  // prologue: kick off staging of t = T-1 into buffer 0
  stage_step(L + 0, (size_t)b * T_ + (T_ - 1));

  float dV0 = 0.f, dV1 = 0.f, werr = 0.f;
  const int mA = lane & 15;          // M (or N) index for WMMA operands
  const int kh = (lane >> 4) << 1;   // K sub-offset per lane half
  int pb = 0;                        // current staging buffer parity

  for (int t = T_ - 1; t >= 0; --t) {
    const size_t bt = (size_t)b * T_ + t;
    const float* I = L + pb * BUFSZ;   // current step's inputs

#if USE_ASYNC
    __builtin_amdgcn_s_wait_asynccnt(0);   // this wave's DMA for buffer pb done
#endif
    __syncthreads();                        // all waves' data + carry visible

    // kick off DMA for step t-1 into the other buffer; overlaps this step
    if (t > 0) stage_step(L + (pb ^ 1) * BUFSZ, bt - 1);

    // register-cache Vx (hot in the einsums)
    float vxr[NS];
#pragma unroll
    for (int sx = 0; sx < NS; ++sx) vxr[sx] = L[LVX + sx];

    // ---- P2: Qx/Qu, r/cinv/sc; wave 0: WMMA chain 1  W = [fx|fu]^T * Vxx ----
    for (int i = tid; i < 16; i += NT) {
      if (i < NS) {
        float acc = I[RQX + i];
        for (int j = 0; j < MM; ++j) acc += I[RCX + j * NS + i] * I[RS + j];
        for (int k = 0; k < NS; ++k) acc += I[RFX + k * NS + i] * vxr[k];
        L[LQXV + i] = acc;
      } else {
        const int a = i - NS;
        float acc = I[RQU + a];
        for (int j = 0; j < MM; ++j) acc += I[RCU + j * NC + a] * I[RS + j];
        for (int k = 0; k < NS; ++k) acc += I[RFU + k * NC + a] * vxr[k];
        L[LQUV + a] = acc;
      }
    }
    for (int j = tid; j < MM; j += NT) {
      const float cj = I[RC + j], sj = I[RS + j];
      L[LRR + j]   = sj * cj + MU;
      L[LCINV + j] = 1.0f / cj;
      L[LSC + j]   = sj / cj;
    }
    if (wv == 0) {
      v8f wacc = {0.f, 0.f, 0.f, 0.f, 0.f, 0.f, 0.f, 0.f};
#pragma unroll
      for (int kc = 0; kc < 3; ++kc) {
        const int k0 = 4 * kc + kh, k1 = k0 + 1;
        v2f av, bv;
        av.x = (mA < NS) ? I[RFX + k0 * NS + mA] : I[RFU + k0 * NC + (mA - NS)];
        av.y = (mA < NS) ? I[RFX + k1 * NS + mA] : I[RFU + k1 * NC + (mA - NS)];
        bv.x = L[LVXX + k0 * 16 + mA];
        bv.y = L[LVXX + k1 * 16 + mA];
        wacc = __builtin_amdgcn_wmma_f32_16x16x4_f32(false, av, false, bv,
                                                     (short)0, wacc, false, false);
      }
#pragma unroll
      for (int r = 0; r < 8; ++r)
        L[LWT + (r + ((lane >> 4) << 3)) * 16 + mA] = wacc[r];
    }
    __syncthreads();

    // ---- P3: wave 0: WMMA chain 2  Z = W * [fx|fu];  wave 1: cuSCcu ----
    if (wv == 0) {
      v8f zacc = {0.f, 0.f, 0.f, 0.f, 0.f, 0.f, 0.f, 0.f};
#pragma unroll
      for (int kc = 0; kc < 3; ++kc) {
        const int k0 = 4 * kc + kh, k1 = k0 + 1;
        v2f av, bv;
        av.x = L[LWT + mA * 16 + k0];
        av.y = L[LWT + mA * 16 + k1];
        bv.x = (mA < NS) ? I[RFX + k0 * NS + mA] : I[RFU + k0 * NC + (mA - NS)];
        bv.y = (mA < NS) ? I[RFX + k1 * NS + mA] : I[RFU + k1 * NC + (mA - NS)];
        zacc = __builtin_amdgcn_wmma_f32_16x16x4_f32(false, av, false, bv,
                                                     (short)0, zacc, false, false);
      }
#pragma unroll
      for (int r = 0; r < 8; ++r)
        L[LZT + (r + ((lane >> 4) << 3)) * 16 + mA] = zacc[r];
    }
    if (wv == 1) {
      for (int e = lane; e < NC * NC; e += 32) {
        const int a = e >> 2, bb = e & 3;
        float acc = 0.f;
        for (int j = 0; j < MM; ++j)
          acc += I[RCU + j * NC + a] * L[LSC + j] * I[RCU + j * NC + bb];
        L[LCUSC + e] = acc;
      }
    }
    __syncthreads();

    // ---- P4: Qxx/Qxu/Quu (Z tile + v.fxx / v.fxu / v.fuu einsums) ----
    for (int e = tid; e < NS * NS; e += NT) {
      const int i = e / NS, j = e % NS;
      float acc = I[RQXX + e] + L[LZT + i * 16 + j];
#pragma unroll
      for (int sx = 0; sx < NS; ++sx) acc += vxr[sx] * I[RFXX + sx * NS * NS + e];
      L[LQXXA + e] = acc;
    }
    for (int e = tid; e < NS * NC; e += NT) {
      const int i = e >> 2, j = e & 3;
      float acc = I[RQXU + e] + L[LZT + i * 16 + 12 + j];
#pragma unroll
      for (int sx = 0; sx < NS; ++sx) acc += vxr[sx] * I[RFXU + sx * NS * NC + e];
      L[LQXUA + e] = acc;
    }
    for (int e = tid; e < NC * NC; e += NT) {
      const int i = e >> 2, j = e & 3;
      float acc = I[RQUU + e] + L[LZT + (12 + i) * 16 + 12 + j];
#pragma unroll
      for (int sx = 0; sx < NS; ++sx) acc += vxr[sx] * I[RFUU + sx * NC * NC + e];
      L[LQUUR + e] = acc;
    }
    __syncthreads();

    // ---- P5: symmetrize Quu, G, Quu2, Qu2, tempQux, Qxx2, Qx2 ----
    for (int e = tid; e < NC * NC; e += NT) {
      const int i = e >> 2, j = e & 3;
      const float qs = 0.5f * (L[LQUUR + e] + L[LQUUR + j * 4 + i]);
      L[LQUUS + e] = qs;
      L[LG + e]    = qs + I[RQUU + e] * REG_FAC - L[LCUSC + e];
      L[LQUU2 + e] = qs - L[LCUSC + e];
    }
    for (int a = tid; a < NC; a += NT) {
      float acc = L[LQUV + a];
      for (int j = 0; j < MM; ++j)
        acc -= I[RCU + j * NC + a] * L[LCINV + j] * L[LRR + j];
      L[LQU2 + a] = acc;
    }
    for (int e = tid; e < NC * NS; e += NT) {
      const int a = e / NS, i = e % NS;
      float acc = L[LQXUA + i * NC + a];
      for (int j = 0; j < MM; ++j)
        acc -= I[RCU + j * NC + a] * L[LSC + j] * I[RCX + j * NS + i];
      L[LTQUX + e] = acc;
    }
    for (int e = tid; e < NS * NS; e += NT) {
      const int i = e / NS, j = e % NS;
      float acc = L[LQXXA + e];
      for (int m = 0; m < MM; ++m)
        acc -= I[RCX + m * NS + i] * L[LSC + m] * I[RCX + m * NS + j];
      L[LQXX2 + e] = acc;
    }
    for (int i = tid; i < NS; i += NT) {
      float acc = L[LQXV + i];
      for (int m = 0; m < MM; ++m)
        acc -= I[RCX + m * NS + i] * L[LCINV + m] * L[LRR + m];
      L[LQX2 + i] = acc;
    }
    __syncthreads();

    // ---- P6: 4x4 SPD solve, one RHS column per thread (13 cols) ----
    {
      float A4[4][4], xr[4];
#pragma unroll
      for (int i = 0; i < 4; ++i)
#pragma unroll
        for (int j = 0; j < 4; ++j) A4[i][j] = L[LG + i * 4 + j];
      const int col = tid;
      const int ci = (col >= 1 && col <= 12) ? (col - 1) : 0;
#pragma unroll
      for (int a = 0; a < 4; ++a)
        xr[a] = (col == 0) ? L[LQU2 + a] : L[LTQUX + a * NS + ci];
#pragma unroll
      for (int k = 0; k < 3; ++k) {
        const float inv = 1.0f / A4[k][k];
#pragma unroll
        for (int i = k + 1; i < 4; ++i) {
          const float f = A4[i][k] * inv;
#pragma unroll
          for (int j = k + 1; j < 4; ++j) A4[i][j] -= f * A4[k][j];
          xr[i] -= f * xr[k];
        }
      }
#pragma unroll
      for (int i = 3; i >= 0; --i) {
        float acc = xr[i];
#pragma unroll
        for (int j = i + 1; j < 4; ++j) acc -= A4[i][j] * xr[j];
        xr[i] = acc / A4[i][i];
      }
      if (tid < 13) {
#pragma unroll
        for (int a = 0; a < 4; ++a) L[LKK + a * 16 + tid] = -xr[a];
      }
    }
    __syncthreads();

    // ---- P7: outputs ku/Ku/ks/Ks, KtQ, QxuKu, dV, errors ----
    float* gku = out + O_KU + bt * NC;
    for (int a = tid; a < NC; a += NT) gku[a] = L[LKK + a * 16];
    float* gKu = out + O_KUM + bt * NC * NS;
    for (int e = tid; e < NC * NS; e += NT) {
      const int a = e / NS, i = e % NS;
      gKu[e] = L[LKK + a * 16 + 1 + i];
    }
    float* gks = out + O_KS + bt * MM;
    for (int j = tid; j < MM; j += NT) {
      float cuku = 0.f;
      for (int a = 0; a < NC; ++a) cuku += I[RCU + j * NC + a] * L[LKK + a * 16];
      gks[j] = -L[LCINV + j] * (L[LRR + j] + I[RS + j] * cuku);
    }
    float* gKs = out + O_KSM + bt * MM * NS;
    for (int e = tid; e < MM * NS; e += NT) {
      const int j = e / NS, i = e % NS;
      float acc = I[RCX + j * NS + i];
      for (int a = 0; a < NC; ++a) acc += I[RCU + j * NC + a] * L[LKK + a * 16 + 1 + i];
      gKs[e] = -(L[LSC + j] * acc);
    }
    for (int e = tid; e < NS * NC; e += NT) {
      const int i = e >> 2, a = e & 3;
      float acc = 0.f;
      for (int bb = 0; bb < 4; ++bb)
        acc += L[LKK + bb * 16 + 1 + i] * L[LQUU2 + bb * 4 + a];
      L[LKTQ + e] = acc;
    }
    for (int e = tid; e < NS * NS; e += NT) {
      const int i = e / NS, j = e % NS;
      float acc = 0.f;
      for (int a = 0; a < NC; ++a)
        acc += L[LTQUX + a * NS + i] * L[LKK + a * 16 + 1 + j];
      L[LQK + e] = acc;
    }
    {
      float d0 = 0.f, d1 = 0.f;
      for (int a = 0; a < NC; ++a) d0 += L[LKK + a * 16] * L[LQU2 + a];
      for (int a = 0; a < NC; ++a)
        for (int bb = 0; bb < NC; ++bb)
          d1 += L[LKK + a * 16] * L[LQUU2 + a * 4 + bb] * L[LKK + bb * 16];
      dV0 += d0;
      dV1 += 0.5f * d1;
      float qe = 0.f, me = 0.f;
      for (int a = 0; a < NC; ++a) qe = fmaxf(qe, fabsf(L[LQU2 + a]));
      for (int j = 0; j < MM; ++j) me = fmaxf(me, fabsf(L[LRR + j]));
      werr = fmaxf(werr, fmaxf(qe, me));
    }
    __syncthreads();

    // ---- P8: Vx_new and unsymmetrized Vxx_new ----
    for (int i = tid; i < NS; i += NT) {
      float acc = L[LQX2 + i];
      for (int a = 0; a < NC; ++a) acc += L[LKK + a * 16 + 1 + i] * L[LQU2 + a];
      for (int a = 0; a < NC; ++a) acc += L[LKTQ + i * 4 + a] * L[LKK + a * 16];
      for (int a = 0; a < NC; ++a) acc += L[LTQUX + a * NS + i] * L[LKK + a * 16];
      L[LVXN + i] = acc;
    }
    for (int e = tid; e < NS * NS; e += NT) {
      const int i = e / NS, j = e % NS;
      float acc = L[LQXX2 + e] + L[LQK + j * NS + i] + L[LQK + e];
      for (int a = 0; a < NC; ++a)
        acc += L[LKTQ + i * 4 + a] * L[LKK + a * 16 + 1 + j];
      L[LVN + e] = acc;
    }
    __syncthreads();

    // ---- P9: commit symmetrized carry ----
    for (int e = tid; e < NS * NS; e += NT) {
      const int i = e / NS, j = e % NS;
      L[LVXX + i * 16 + j] = 0.5f * (L[LVN + e] + L[LVN + j * NS + i]);
    }
    for (int i = tid; i < NS; i += NT) L[LVX + i] = L[LVXN + i];
    __syncthreads();

    pb ^= 1;
  }

  // ---- final per-batch outputs ----
  for (int i = tid; i < NS; i += NT) out[O_VX + (size_t)b * NS + i] = L[LVX + i];
  for (int e = tid; e < NS * NS; e += NT)
    out[O_VXX + (size_t)b * NS * NS + e] = L[LVXX + (e / NS) * 16 + (e % NS)];
  if (tid == 0) {
    out[O_DV + (size_t)b * 2 + 0] = dV0;
    out[O_DV + (size_t)b * 2 + 1] = dV1;
    // nonnegative floats: uint bit-pattern max == float max
    atomicMax(reinterpret_cast<unsigned int*>(out + O_OPT), __float_as_uint(werr));
  }
}

extern "C" void kernel_launch(void* const* d_in, const int* in_sizes, int n_in,
                              void* d_out, int out_size, void* d_ws, size_t ws_size,
                              hipStream_t stream) {
  (void)in_sizes; (void)n_in; (void)out_size; (void)d_ws; (void)ws_size;
  const float* qx  = (const float*)d_in[0];
  const float* qu  = (const float*)d_in[1];
  const float* qxx = (const float*)d_in[2];
  const float* qxu = (const float*)d_in[3];
  const float* quu = (const float*)d_in[4];
  const float* fx  = (const float*)d_in[5];
  const float* fu  = (const float*)d_in[6];
  const float* fxx = (const float*)d_in[7];
  const float* fxu = (const float*)d_in[8];
  const float* fuu = (const float*)d_in[9];
  const float* cx  = (const float*)d_in[10];
  const float* cu  = (const float*)d_in[11];
  const float* c   = (const float*)d_in[12];
  const float* s   = (const float*)d_in[13];
  const float* px  = (const float*)d_in[14];
  const float* pxx = (const float*)d_in[15];
  float* out = (float*)d_out;

  ddp_init_kernel<<<1, 1, 0, stream>>>(out);
  ddp_bwd_kernel<<<dim3(B_), dim3(NT), 0, stream>>>(
      qx, qu, qxx, qxu, quu, fx, fu, fxx, fxu, fuu, cx, cu, c, s, px, pxx, out);
}